// SequenceRouter_5660766896432
// MI455X (gfx1250) — compile-verified
//
#include <hip/hip_runtime.h>
#include <hip/hip_bf16.h>

// ---------------------------------------------------------------------------
// SequenceRouter fused kernel for MI455X (gfx1250, wave32).
//
// Roofline: stream r_pooled (268 MB) once -> ~12 us floor at 23.3 TB/s.
// fp32 kept end-to-end (top-k ordering is precision sensitive) via
// V_WMMA_F32_16X16X4_F32. Data movement uses the CDNA5 async path:
// GLOBAL_LOAD_ASYNC_TO_LDS_B128 + ASYNCcnt, double-buffered so chunk i+1's
// HBM stream overlaps chunk i's WMMA compute. Fused: GEMM1 -> SiLU ->
// GEMM2 -> top-8 softmax scatter per 64-row block.
// ---------------------------------------------------------------------------

typedef float v2f __attribute__((ext_vector_type(2)));
typedef float v4f __attribute__((ext_vector_type(4)));
typedef float v8f __attribute__((ext_vector_type(8)));

#define NB   16384   // batch
#define ND   4096    // main feature dim; +3 tail features in epilogue
#define NE   64      // experts
#define NH   128     // hidden
#define ROWS 64      // batch rows per block
#define KC   64      // K chunk for GEMM1
#define NCH  (ND / KC)

#define A_STR 68     // A row stride (floats): bank stride 4, 272B rows (16B aligned)
#define BS1   144    // W1 row stride (floats): 2*144 % 64 == 32 -> halves on disjoint banks
#define BS2   80     // W2 row stride (floats): 2*80  % 64 == 32
#define H_STR 132
#define L_STR 68

#define ABUF (ROWS * A_STR)   // floats per A buffer
#define BBUF (KC * BS1)       // floats per B buffer

// generic shared pointer -> wave-relative LDS byte address (low 32 bits)
#define LDS_U32(p) ((uint32_t)(uintptr_t)(const void*)(p))

__device__ __forceinline__ void async_b128(uint32_t lds, int32_t goff, const void* base)
{
    // GVS mode: SGPR64 base + per-lane VGPR byte offset; 16B per lane -> 512B/wave
    asm volatile("global_load_async_to_lds_b128 %0, %1, %2"
                 :: "v"(lds), "v"(goff), "s"(base) : "memory");
}

#define WAIT_ASYNC_12() asm volatile("s_wait_asynccnt 0xc" ::: "memory")
#define WAIT_ASYNC_0()  asm volatile("s_wait_asynccnt 0x0" ::: "memory")

__global__ __launch_bounds__(256)
void router_fused(const float* __restrict__ r_pooled,
                  const float* __restrict__ step_frac,
                  const float* __restrict__ hidden_norm,
                  const float* __restrict__ confidence,
                  const float* __restrict__ W1,
                  const float* __restrict__ b1,
                  const float* __restrict__ W2,
                  const float* __restrict__ b2,
                  float* __restrict__ out_w,
                  float* __restrict__ out_lg)
{
    __shared__ float A_lds[2 * ABUF];      // 34.8 KB, double buffered
    __shared__ float B_lds[2 * BBUF];      // 73.7 KB, double buffered (W2 reuses it)
    __shared__ float h_lds[ROWS * H_STR];  // 33.8 KB
    __shared__ float lg_lds[ROWS * L_STR]; // 17.4 KB
    __shared__ float hn_s[ROWS];
    __shared__ float cf_s[ROWS];

    const int tid  = threadIdx.x;
    const int lane = tid & 31;
    const int wv   = tid >> 5;        // 8 waves
    const int l15  = lane & 15;
    const int lh   = lane >> 4;       // half-wave select
    const int mi   = wv >> 1;         // M tile (0..3)
    const int nb   = (wv & 1) * 64;   // N half for GEMM1
    const int rowBase = blockIdx.x * ROWS;

    if (tid < ROWS) {
        hn_s[tid] = hidden_norm[rowBase + tid];
        cf_s[tid] = confidence[rowBase + tid];
    }
    const float sf = step_frac[0];

    const uint32_t aBase = LDS_U32(A_lds);
    const uint32_t bBase = LDS_U32(B_lds);

    // Issue one chunk's async copies: 4 (A) + 8 (B) B128 ops per wave.
    auto stage = [&](int kc, int buf) {
        #pragma unroll
        for (int i = 0; i < 4; ++i) {          // A: 2 rows per op (32 lanes x 16B)
            int row = 8 * wv + 2 * i + lh;
            int32_t g = (int32_t)(((size_t)(rowBase + row) * ND + kc + l15 * 4) * 4);
            uint32_t l = aBase + (uint32_t)((buf * ABUF + row * A_STR + l15 * 4) * 4);
            async_b128(l, g, r_pooled);
        }
        #pragma unroll
        for (int i = 0; i < 8; ++i) {          // B: 1 k-row (512B) per op
            int kr = 8 * wv + i;
            int32_t g = (int32_t)((((size_t)(kc + kr)) * NH + lane * 4) * 4);
            uint32_t l = bBase + (uint32_t)((buf * BBUF + kr * BS1 + lane * 4) * 4);
            async_b128(l, g, W1);
        }
    };

    // ---------------- GEMM1: h = features @ W1 -------------------------------
    v8f acc[4] = {};
    stage(0, 0);

    for (int c = 0; c < NCH; ++c) {
        if (c + 1 < NCH) { stage((c + 1) * KC, (c + 1) & 1); WAIT_ASYNC_12(); }
        else             { WAIT_ASYNC_0(); }
        __syncthreads();   // chunk c resident in all waves' view

        const float* Ab = A_lds + (c & 1) * ABUF;
        const float* Bb = B_lds + (c & 1) * BBUF;

        #pragma unroll 4
        for (int kk = 0; kk < KC; kk += 4) {
            // A frag (ISA 16x4 layout): lanes 0-15 K=kk,kk+1; lanes 16-31 K=kk+2,kk+3
            v2f a = *(const v2f*)(Ab + (mi * 16 + l15) * A_STR + kk + 2 * lh);
            const float* bc = Bb + (kk + 2 * lh) * BS1 + nb + l15;
            #pragma unroll
            for (int nt = 0; nt < 4; ++nt) {
                v2f b;                         // merges into ds_load_2addr_b32
                b.x = bc[nt * 16];
                b.y = bc[BS1 + nt * 16];
                acc[nt] = __builtin_amdgcn_wmma_f32_16x16x4_f32(
                    false, a, false, b, (short)0, acc[nt], false, false);
            }
        }
        __syncthreads();   // before buffer (c&1) is overwritten by stage(c+2)
    }

    // Stage W2 (128x64) async into the freed B buffer, overlapped with epilogue.
    #pragma unroll
    for (int i = 0; i < 8; ++i) {              // 2 rows per op
        int row = 16 * wv + 2 * i + lh;
        int32_t g = (int32_t)((row * NE + l15 * 4) * 4);
        uint32_t l = bBase + (uint32_t)((row * BS2 + l15 * 4) * 4);
        async_b128(l, g, W2);
    }

    // Epilogue 1: tail features (W1 rows 4096..4098) + b1, SiLU, h -> LDS.
    #pragma unroll
    for (int nt = 0; nt < 4; ++nt) {
        int col = nb + nt * 16 + l15;
        float w96 = W1[(size_t)4096 * NH + col];
        float w97 = W1[(size_t)4097 * NH + col];
        float w98 = W1[(size_t)4098 * NH + col];
        float base = sf * w96 + b1[col];
        #pragma unroll
        for (int v = 0; v < 8; ++v) {
            int row = mi * 16 + v + lh * 8;    // C/D layout: vgpr v -> M=v / v+8
            float x = acc[nt][v] + base + hn_s[row] * w97 + cf_s[row] * w98;
            float s = x * __frcp_rn(1.0f + __expf(-x));   // silu
            h_lds[row * H_STR + col] = s;
        }
    }
    WAIT_ASYNC_0();
    __syncthreads();

    // ---------------- GEMM2: logits = h @ W2 + b2 ----------------------------
    v8f acc2[2] = {};
    const int ntb = (wv & 1) * 2;
    #pragma unroll 4
    for (int kk = 0; kk < NH; kk += 4) {
        v2f a = *(const v2f*)(h_lds + (mi * 16 + l15) * H_STR + kk + 2 * lh);
        const float* wc = B_lds + (kk + 2 * lh) * BS2 + l15;
        #pragma unroll
        for (int j = 0; j < 2; ++j) {
            v2f b;
            b.x = wc[(ntb + j) * 16];
            b.y = wc[BS2 + (ntb + j) * 16];
            acc2[j] = __builtin_amdgcn_wmma_f32_16x16x4_f32(
                false, a, false, b, (short)0, acc2[j], false, false);
        }
    }

    #pragma unroll
    for (int j = 0; j < 2; ++j) {
        int col = (ntb + j) * 16 + l15;
        float bias = b2[col];
        #pragma unroll
        for (int v = 0; v < 8; ++v) {
            int row = mi * 16 + v + lh * 8;
            float lg = acc2[j][v] + bias;
            lg_lds[row * L_STR + col] = lg;
            out_lg[(size_t)(rowBase + row) * NE + col] = lg;
        }
    }
    __syncthreads();

    // ---------------- Top-8 + softmax + scatter ------------------------------
    if (tid < ROWS) {
        float tv[8]; int ti[8];
        #pragma unroll
        for (int i = 0; i < 8; ++i) { tv[i] = -3.4e38f; ti[i] = 0; }
        for (int j = 0; j < NE; ++j) {
            float xv = lg_lds[tid * L_STR + j];
            int   xi = j;
            #pragma unroll
            for (int i = 0; i < 8; ++i) {      // register insertion network
                bool  gt = xv > tv[i];         // strict: ties keep lower index
                float nv = gt ? xv : tv[i];
                int   ni = gt ? xi : ti[i];
                float ov = tv[i]; int oi = ti[i];
                tv[i] = nv; ti[i] = ni;
                xv = gt ? ov : xv;
                xi = gt ? oi : xi;
            }
        }
        float m = tv[0], s = 0.0f, ex[8];
        #pragma unroll
        for (int i = 0; i < 8; ++i) { ex[i] = __expf(tv[i] - m); s += ex[i]; }
        float inv = __frcp_rn(s);
        float* wrow = A_lds + tid * A_STR;     // reuse A buffer 0 as weights tile
        for (int j = 0; j < NE; ++j) wrow[j] = 0.0f;
        #pragma unroll
        for (int i = 0; i < 8; ++i) wrow[ti[i]] = ex[i] * inv;
    }
    __syncthreads();

    // Coalesced float4 store of the dense weights matrix.
    #pragma unroll
    for (int i = 0; i < 4; ++i) {
        int q = tid + i * 256;
        int r = q >> 4, c4 = (q & 15) << 2;
        v4f v = *(const v4f*)(A_lds + r * A_STR + c4);
        *(v4f*)(out_w + (size_t)(rowBase + r) * NE + c4) = v;
    }
}

extern "C" void kernel_launch(void* const* d_in, const int* in_sizes, int n_in,
                              void* d_out, int out_size, void* d_ws, size_t ws_size,
                              hipStream_t stream) {
    const float* r_pooled    = (const float*)d_in[0];
    const float* step_frac   = (const float*)d_in[1];
    const float* hidden_norm = (const float*)d_in[2];
    const float* confidence  = (const float*)d_in[3];
    const float* W1          = (const float*)d_in[4];
    const float* b1          = (const float*)d_in[5];
    const float* W2          = (const float*)d_in[6];
    const float* b2          = (const float*)d_in[7];
    float* out_w  = (float*)d_out;                        // weights (B*E)
    float* out_lg = (float*)d_out + (size_t)NB * NE;      // logits  (B*E)

    router_fused<<<dim3(NB / ROWS), dim3(256), 0, stream>>>(
        r_pooled, step_frac, hidden_norm, confidence, W1, b1, W2, b2,
        out_w, out_lg);
}